// LocalGrouper_58145267253416
// MI455X (gfx1250) — compile-verified
//
#include <hip/hip_runtime.h>
#include <hip/hip_bf16.h>

typedef __attribute__((ext_vector_type(2))) float v2f;
typedef __attribute__((ext_vector_type(8))) float v8f;

#define BN   16      // batches
#define NP   4096    // points per batch
#define DCH  64      // feature channels
#define SC   1024    // sampled centroids per batch
#define KN   32      // neighbors
#define TILE 16      // centroids per block
#define CH   66      // D+2 grouped channels
#define EPSN 1e-5f

// ---------- helpers ----------
__device__ __forceinline__ unsigned int fkey(float f) {
    unsigned int u = __float_as_uint(f);
    return (u & 0x80000000u) ? ~u : (u | 0x80000000u);  // order-preserving flip
}

__device__ __forceinline__ unsigned long long shflxor_u64(unsigned long long v, int m) {
    unsigned int lo = (unsigned int)v, hi = (unsigned int)(v >> 32);
    lo = __shfl_xor(lo, m, 32);
    hi = __shfl_xor(hi, m, 32);
    return ((unsigned long long)hi << 32) | lo;
}

__device__ __forceinline__ unsigned long long chunk_min(const float* drow, int lane) {
    unsigned long long lm = ~0ull;
    for (int j = lane; j < NP; j += 32) {
        unsigned long long pk = ((unsigned long long)fkey(drow[j]) << 32) | (unsigned int)j;
        if (pk < lm) lm = pk;
    }
    return lm;
}

// ---------- kernel 1: WMMA distances + exact top-32 + mean + ssq partials ----------
__global__ void knn_stats_kernel(const float* __restrict__ xy,
                                 const float* __restrict__ points,
                                 const int*   __restrict__ sample_idx,
                                 int*   __restrict__ idx_buf,      // [BN,SC,KN]
                                 float* __restrict__ mean_buf,     // [BN,SC,CH]
                                 float* __restrict__ ssq_partial)  // [BN*SC/TILE]
{
    extern __shared__ char smem[];
    float* dist  = (float*)smem;                               // TILE*NP f32 (256KB)
    int*   knn   = (int*)(smem + (size_t)TILE * NP * 4);       // TILE*KN
    float* meanl = (float*)((char*)knn + TILE * KN * 4);       // TILE*CH
    float* red   = meanl + TILE * CH;                          // 8

    const int tilesPerBatch = SC / TILE;                       // 64
    const int b    = blockIdx.x / tilesPerBatch;
    const int tile = blockIdx.x % tilesPerBatch;
    const int s0   = tile * TILE;

    const int tid  = threadIdx.x;
    const int lane = tid & 31;
    const int wave = tid >> 5;
    const int half = lane >> 4;
    const int l16  = lane & 15;

    // ---- A fragment: row m=l16 is [-2cx, -2cy, 1, |c|^2]; lanes 0-15 hold K0/K1, 16-31 K2/K3
    const int sj0 = sample_idx[b * SC + s0 + l16];
    const float cx = xy[(b * NP + sj0) * 2 + 0];
    const float cy = xy[(b * NP + sj0) * 2 + 1];
    v2f a;
    a.x = half ? 1.0f                : -2.0f * cx;
    a.y = half ? (cx * cx + cy * cy) : -2.0f * cy;

    // ---- distance tiles: wave w covers points [w*512, w*512+512)
    for (int t = 0; t < 32; ++t) {
        const int p0 = wave * 512 + t * 16;
        const float px = xy[(b * NP + p0 + l16) * 2 + 0];
        const float py = xy[(b * NP + p0 + l16) * 2 + 1];
        v2f bb;                                   // col n: [px, py, |p|^2, 1]
        bb.x = half ? (px * px + py * py) : px;
        bb.y = half ? 1.0f                : py;
        v8f c = {};
        v8f d = __builtin_amdgcn_wmma_f32_16x16x4_f32(false, a, false, bb,
                                                      (short)0, c, false, false);
#pragma unroll
        for (int r = 0; r < 8; ++r) {
            const int m = r + 8 * half;           // C/D layout: VGPR r -> M=r (lo), r+8 (hi)
            dist[m * NP + p0 + l16] = d[r];
        }
    }
    __syncthreads();

    // ---- exact stable top-32 per centroid row (2 rows per wave)
    for (int mi = 0; mi < 2; ++mi) {
        const int m = wave + mi * 8;
        float* drow = dist + (size_t)m * NP;
        unsigned long long lm = chunk_min(drow, lane);
        for (int r = 0; r < KN; ++r) {
            unsigned long long w = lm;
#pragma unroll
            for (int off = 16; off; off >>= 1) {
                unsigned long long o = shflxor_u64(w, off);
                if (o < w) w = o;
            }
            const unsigned int widx = (unsigned int)w;
            if (lane == 0) knn[m * KN + r] = (int)widx;
            if ((widx & 31u) == (unsigned int)lane) {   // winner owns this chunk slot
                drow[widx] = __builtin_inff();
                lm = chunk_min(drow, lane);
            }
        }
    }
    __syncthreads();

    // ---- write neighbor indices
    for (int t = tid; t < TILE * KN; t += blockDim.x) {
        const int g = t / KN, k = t % KN;
        idx_buf[(b * SC + s0 + g) * KN + k] = knn[g * KN + k];
    }

    // ---- per-group mean over K for each of 66 channels
    for (int t = tid; t < TILE * CH; t += blockDim.x) {
        const int g = t / CH, c = t % CH;
        float sum = 0.0f;
        for (int k = 0; k < KN; ++k) {
            const int ni = knn[g * KN + k];
            const float v = (c < DCH) ? points[((size_t)(b * NP + ni)) * DCH + c]
                                      : xy[(b * NP + ni) * 2 + (c - DCH)];
            sum += v;
        }
        const float mu = sum * (1.0f / (float)KN);
        meanl[t] = mu;
        mean_buf[(b * SC + s0 + g) * CH + c] = mu;
    }
    __syncthreads();

    // ---- per-block partial sum of squared deviations (deterministic)
    float acc = 0.0f;
    for (int e = tid; e < TILE * KN * CH; e += blockDim.x) {
        const int g   = e / (KN * CH);
        const int rem = e % (KN * CH);
        const int k = rem / CH, c = rem % CH;
        const int ni = knn[g * KN + k];
        const float v = (c < DCH) ? points[((size_t)(b * NP + ni)) * DCH + c]
                                  : xy[(b * NP + ni) * 2 + (c - DCH)];
        const float dv = v - meanl[g * CH + c];
        acc += dv * dv;
    }
#pragma unroll
    for (int off = 16; off; off >>= 1) acc += __shfl_xor(acc, off, 32);
    if (lane == 0) red[wave] = acc;
    __syncthreads();
    if (tid == 0) {
        float s = 0.0f;
        for (int i = 0; i < 8; ++i) s += red[i];
        ssq_partial[blockIdx.x] = s;
    }
}

// ---------- kernel 2: per-batch std (ddof=1) ----------
__global__ void std_kernel(const float* __restrict__ ssq_partial,
                           float* __restrict__ std_buf)
{
    const int b = threadIdx.x;
    if (b < BN) {
        float s = 0.0f;
        for (int i = 0; i < SC / TILE; ++i) s += ssq_partial[b * (SC / TILE) + i];
        std_buf[b] = sqrtf(s / (float)((size_t)SC * KN * CH - 1));
    }
}

// ---------- kernel 3: new_xy gather ----------
__global__ void newxy_kernel(const float* __restrict__ xy,
                             const int* __restrict__ sample_idx,
                             float* __restrict__ out)
{
    const int t = blockIdx.x * blockDim.x + threadIdx.x;   // BN*SC
    if (t < BN * SC) {
        const int b = t >> 10;
        const int sj = sample_idx[t];
        out[t * 2 + 0] = xy[(b * NP + sj) * 2 + 0];
        out[t * 2 + 1] = xy[(b * NP + sj) * 2 + 1];
    }
}

// ---------- kernel 4: normalized/affine grouped features + anchor ----------
__global__ void assemble_kernel(const float* __restrict__ xy,
                                const float* __restrict__ points,
                                const float* __restrict__ alpha,
                                const float* __restrict__ beta,
                                const int*   __restrict__ sample_idx,
                                const int*   __restrict__ idx_buf,
                                const float* __restrict__ mean_buf,
                                const float* __restrict__ std_buf,
                                float* __restrict__ outp)
{
    const long long e = (long long)blockIdx.x * blockDim.x + threadIdx.x;
    const long long total = (long long)BN * SC * KN * (2 * DCH + 2);
    if (e >= total) return;
    const int c = (int)(e % (2 * DCH + 2));
    long long q = e / (2 * DCH + 2);
    const int k = (int)(q % KN); q /= KN;
    const int s = (int)(q % SC); q /= SC;
    const int b = (int)q;

    float v;
    if (c < CH) {
        const int ni = idx_buf[(b * SC + s) * KN + k];
        const float raw = (c < DCH) ? points[((size_t)(b * NP + ni)) * DCH + c]
                                    : xy[(b * NP + ni) * 2 + (c - DCH)];
        const float mu = mean_buf[(b * SC + s) * CH + c];
        const float nv = (raw - mu) / (std_buf[b] + EPSN);
        v = alpha[c] * nv + beta[c];
    } else {
        const int sj = sample_idx[b * SC + s];
        v = points[((size_t)(b * NP + sj)) * DCH + (c - CH)];
    }
    outp[e] = v;
}

extern "C" void kernel_launch(void* const* d_in, const int* in_sizes, int n_in,
                              void* d_out, int out_size, void* d_ws, size_t ws_size,
                              hipStream_t stream) {
    const float* xy     = (const float*)d_in[0];
    const float* points = (const float*)d_in[1];
    const float* alpha  = (const float*)d_in[2];
    const float* beta   = (const float*)d_in[3];
    const int*   sidx   = (const int*)d_in[4];
    float* out = (float*)d_out;

    // workspace layout
    char* ws = (char*)d_ws;
    int*   idx_buf     = (int*)ws;                                       // BN*SC*KN
    float* mean_buf    = (float*)(ws + (size_t)BN * SC * KN * 4);        // BN*SC*CH
    float* ssq_partial = (float*)((char*)mean_buf + (size_t)BN * SC * CH * 4); // BN*64
    float* std_buf     = ssq_partial + BN * (SC / TILE);                 // BN

    const size_t shbytes = (size_t)TILE * NP * 4 + TILE * KN * 4 + TILE * CH * 4 + 8 * 4;
    (void)hipFuncSetAttribute((const void*)knn_stats_kernel,
                              hipFuncAttributeMaxDynamicSharedMemorySize, (int)shbytes);

    knn_stats_kernel<<<BN * (SC / TILE), 256, shbytes, stream>>>(
        xy, points, sidx, idx_buf, mean_buf, ssq_partial);

    std_kernel<<<1, 32, 0, stream>>>(ssq_partial, std_buf);

    newxy_kernel<<<(BN * SC + 255) / 256, 256, 0, stream>>>(xy, sidx, out);

    const long long total = (long long)BN * SC * KN * (2 * DCH + 2);
    assemble_kernel<<<(unsigned int)((total + 255) / 256), 256, 0, stream>>>(
        xy, points, alpha, beta, sidx, idx_buf, mean_buf, std_buf,
        out + (size_t)BN * SC * 2);
}